// QuantumEmulator_65481071404117
// MI455X (gfx1250) — compile-verified
//
#include <hip/hip_runtime.h>

typedef __attribute__((ext_vector_type(2))) float v2f;
typedef __attribute__((ext_vector_type(4))) float v4f;
typedef __attribute__((ext_vector_type(8))) float v8f;

#define IN_F   784
#define OUT_F  4
#define NCHAIN 49

#define WAVES_PER_BLOCK 8
#define ROWS_PER_WAVE   32          // two 16-row WMMA tiles per wave
#define ROWS_PER_BLOCK  (WAVES_PER_BLOCK * ROWS_PER_WAVE)   // 256

__global__ __launch_bounds__(256)
void mlp_chain_wmma_f32(const float* __restrict__ x,
                        const float* __restrict__ W0,
                        const float* __restrict__ Ws,
                        float* __restrict__ out)
{
    __shared__ float lds_w0[OUT_F * IN_F];                         // 12544 B
    __shared__ float lds_h[WAVES_PER_BLOCK][ROWS_PER_WAVE][OUT_F]; // 4096 B

    const int tid = threadIdx.x;

    // Stage W0 (row-major [4][784]) into LDS, coalesced.
    for (int i = tid; i < OUT_F * IN_F; i += ROWS_PER_BLOCK)
        lds_w0[i] = W0[i];
    __syncthreads();

    const int wave = tid >> 5;
    const int lane = tid & 31;
    const int m    = lane & 15;     // M (tile row) for A, N (col) for B/C/D
    const int hi   = lane >> 4;     // half-wave select: K offset +2 in A/B frags

    const long rowBase = (long)blockIdx.x * ROWS_PER_BLOCK + (long)wave * ROWS_PER_WAVE;

    // A-fragment streaming pointers for the two 16-row tiles.
    // A 16x4 f32 layout: lane m holds row m; VGPR0/1 = K, K+1 (lanes 0-15)
    // or K+2, K+3 (lanes 16-31)  -> one b64 load per chunk per lane.
    const float* a0 = x + (rowBase +  0 + m) * IN_F + 2 * hi;
    const float* a1 = x + (rowBase + 16 + m) * IN_F + 2 * hi;

    // B-fragment source: B[k][n] = (n < 4) ? W0[n][k] : 0
    const float* brow  = &lds_w0[(m & (OUT_F - 1)) * IN_F];
    const float  bmask = (m < OUT_F) ? 1.0f : 0.0f;

    v8f acc0 = {};
    v8f acc1 = {};

    #pragma unroll 8
    for (int k = 0; k < IN_F; k += 4) {
        if ((k & 31) == 0) {   // one cacheline ahead per 128B consumed
            __builtin_prefetch(a0 + k + 32, 0, 1);
            __builtin_prefetch(a1 + k + 32, 0, 1);
        }
        v2f A0 = *(const v2f*)(a0 + k);
        v2f A1 = *(const v2f*)(a1 + k);
        v2f B;
        B.x = bmask * brow[k + 2 * hi + 0];
        B.y = bmask * brow[k + 2 * hi + 1];
        acc0 = __builtin_amdgcn_wmma_f32_16x16x4_f32(false, A0, false, B,
                                                     (short)0, acc0, false, false);
        acc1 = __builtin_amdgcn_wmma_f32_16x16x4_f32(false, A1, false, B,
                                                     (short)0, acc1, false, false);
    }

    // ReLU + transpose D-layout (row spread across lanes) -> row-per-lane via LDS.
    // D layout: VGPR r <-> M = r + 8*hi, N = m.
    if (m < OUT_F) {
        #pragma unroll
        for (int r = 0; r < 8; ++r) {
            lds_h[wave][r + 8 * hi +  0][m] = fmaxf(acc0[r], 0.0f);
            lds_h[wave][r + 8 * hi + 16][m] = fmaxf(acc1[r], 0.0f);
        }
    }
    __syncthreads();

    // Each lane now owns one batch row (4 features).
    float h0 = lds_h[wave][lane][0];
    float h1 = lds_h[wave][lane][1];
    float h2 = lds_h[wave][lane][2];
    float h3 = lds_h[wave][lane][3];

    // 49 sequential 4x4 matmul+ReLU steps; Ws indices are wave-uniform ->
    // scalar loads through the constant cache.
    for (int s = 0; s < NCHAIN; ++s) {
        const float* W = Ws + s * (OUT_F * OUT_F);
        float t0 = fmaxf(h0 * W[ 0] + h1 * W[ 1] + h2 * W[ 2] + h3 * W[ 3], 0.0f);
        float t1 = fmaxf(h0 * W[ 4] + h1 * W[ 5] + h2 * W[ 6] + h3 * W[ 7], 0.0f);
        float t2 = fmaxf(h0 * W[ 8] + h1 * W[ 9] + h2 * W[10] + h3 * W[11], 0.0f);
        float t3 = fmaxf(h0 * W[12] + h1 * W[13] + h2 * W[14] + h3 * W[15], 0.0f);
        h0 = t0; h1 = t1; h2 = t2; h3 = t3;
    }

    // Coalesced b128 store: 4 contiguous floats per row, one row per lane.
    v4f o = { h0, h1, h2, h3 };
    *(v4f*)(out + (rowBase + lane) * OUT_F) = o;
}

extern "C" void kernel_launch(void* const* d_in, const int* in_sizes, int n_in,
                              void* d_out, int out_size, void* d_ws, size_t ws_size,
                              hipStream_t stream) {
    const float* x  = (const float*)d_in[0];   // [rows, 784]
    const float* W0 = (const float*)d_in[1];   // [4, 784]
    const float* Ws = (const float*)d_in[2];   // [49, 4, 4]
    float* out = (float*)d_out;                // [rows, 4]

    const int rows = in_sizes[0] / IN_F;       // 262144
    const int grid = rows / ROWS_PER_BLOCK;    // 1024 (rows divisible by 256)

    mlp_chain_wmma_f32<<<grid, ROWS_PER_BLOCK, 0, stream>>>(x, W0, Ws, out);
}